// GATModel_12927851561249
// MI455X (gfx1250) — compile-verified
//
#include <hip/hip_runtime.h>
#include <math.h>

// ---------------------------------------------------------------------------
// GAT forward on gfx1250 (wave32, WMMA f32 16x16x4).
// ---------------------------------------------------------------------------

typedef float v2f __attribute__((ext_vector_type(2)));
typedef float v8f __attribute__((ext_vector_type(8)));

#define F_IN 256
#define HHID 64

// ---------------- small utility kernels ----------------

__global__ void fill_f32(float* __restrict__ p, long n, float v) {
    long i = (long)blockIdx.x * blockDim.x + threadIdx.x;
    if (i < n) p[i] = v;
}

// Column-wise sum / sumsq over x[N,256]; 256 threads = one per column.
__global__ void bn_stats(const float* __restrict__ x, float* __restrict__ sum,
                         float* __restrict__ sumsq, int n) {
    int col = threadIdx.x;                       // 0..255
    int rows_per_blk = (n + gridDim.x - 1) / gridDim.x;
    int r0 = blockIdx.x * rows_per_blk;
    int r1 = min(n, r0 + rows_per_blk);
    float s = 0.f, ss = 0.f;
    for (int r = r0; r < r1; ++r) {
        float v = x[(long)r * F_IN + col];
        s += v; ss += v * v;
    }
    atomicAdd(&sum[col], s);
    atomicAdd(&sumsq[col], ss);
}

// xn = x*scale + shift  with scale = gamma*rsqrt(var+eps), shift = beta - mu*scale
__global__ void bn_finalize(const float* __restrict__ sum, const float* __restrict__ sumsq,
                            const float* __restrict__ gamma, const float* __restrict__ beta,
                            float* __restrict__ sc, float* __restrict__ sh, int n) {
    int k = threadIdx.x;                         // 256 threads
    float mu  = sum[k] / (float)n;
    float var = sumsq[k] / (float)n - mu * mu;
    float inv = rsqrtf(var + 1e-5f);
    float s = inv * gamma[k];
    sc[k] = s;
    sh[k] = beta[k] - mu * s;
}

// ---------------- WMMA GEMM: C[M,NCOL] = A'[M,K] @ B[K,NCOL] ----------------
// A' = A*scale+shift per column when scale!=nullptr (BatchNorm fusion).
// One wave per 16x16 output tile; workgroup = strip of 16 rows x NCOL cols.
// blockDim.x must equal 2*NCOL (=32 * NCOL/16 lanes).

template <int K, int NCOL>
__global__ void gemm_wmma(const float* __restrict__ A, const float* __restrict__ B,
                          float* __restrict__ C,
                          const float* __restrict__ scale, const float* __restrict__ shift) {
    __shared__ float Alds[16 * K];
    const int row0 = blockIdx.x * 16;
    const int tid  = threadIdx.x;

    // Cooperative staging of the 16xK A-strip (BN applied on the fly).
    for (int i = tid; i < 16 * K; i += blockDim.x) {
        int r = i / K, k = i % K;
        float v = A[(long)(row0 + r) * K + k];
        if (scale) v = v * scale[k] + shift[k];
        Alds[i] = v;
    }
    __syncthreads();

    const int wave  = tid >> 5;          // which 16-col tile
    const int lane  = tid & 31;
    const int col0  = wave * 16;
    const int lrow  = lane & 15;         // M (A) / N (B) index within tile
    const int khalf = (lane >> 4) * 2;   // lanes 0-15 -> K+0/K+1, lanes 16-31 -> K+2/K+3

    v8f acc = {};
    for (int kk = 0; kk < K; kk += 4) {
        v2f a, b;
        a[0] = Alds[lrow * K + kk + khalf];
        a[1] = Alds[lrow * K + kk + khalf + 1];
        b[0] = B[(long)(kk + khalf)     * NCOL + col0 + lrow];
        b[1] = B[(long)(kk + khalf + 1) * NCOL + col0 + lrow];
        acc = __builtin_amdgcn_wmma_f32_16x16x4_f32(
            /*neg_a=*/false, a, /*neg_b=*/false, b,
            /*c_mod=*/(short)0, acc, /*reuse_a=*/false, /*reuse_b=*/false);
    }

    // C/D layout: VGPR r -> lanes 0-15 M=r, lanes 16-31 M=r+8; N = lane&15.
    const int rbase = (lane >> 4) * 8;
    const int c = col0 + lrow;
    #pragma unroll
    for (int r = 0; r < 8; ++r)
        C[(long)(row0 + rbase + r) * NCOL + c] = acc[r];
}

// ---------------- attention dot products ----------------
// asrc[n,h] = h[n,h,:].a_src[h,:], adst likewise.
__global__ void att_dots(const float* __restrict__ h, const float* __restrict__ a_src,
                         const float* __restrict__ a_dst, float* __restrict__ asrc,
                         float* __restrict__ adst, int n, int heads) {
    int i = blockIdx.x * blockDim.x + threadIdx.x;
    if (i >= n * heads) return;
    int node = i / heads, hd = i % heads;
    const float* hp = h + (long)node * heads * HHID + hd * HHID;
    const float* as = a_src + hd * HHID;
    const float* ad = a_dst + hd * HHID;
    float s = 0.f, d = 0.f;
    #pragma unroll 4
    for (int k = 0; k < HHID; ++k) { float v = hp[k]; s += v * as[k]; d += v * ad[k]; }
    asrc[i] = s;
    adst[i] = d;
}

// float atomic max via signed/unsigned bit ordering trick
__device__ inline void atomicMaxF(float* addr, float val) {
    if (val >= 0.f) atomicMax((int*)addr, __float_as_int(val));
    else            atomicMin((unsigned int*)addr, __float_as_uint(val));
}

__device__ inline float leaky02(float v) { return v > 0.f ? v : 0.2f * v; }

// pass 1: segment max over dst
__global__ void att_edge_max(const int* __restrict__ src, const int* __restrict__ dst,
                             const float* __restrict__ asrc, const float* __restrict__ adst,
                             float* __restrict__ emax, int E, int n, int heads) {
    long i = (long)blockIdx.x * blockDim.x + threadIdx.x;
    long total = (long)(E + n) * heads;
    if (i >= total) return;
    long e = i / heads; int hd = (int)(i % heads);
    int s, d;
    if (e < E) { s = src[e]; d = dst[e]; } else { s = d = (int)(e - E); }
    float v = leaky02(asrc[(long)s * heads + hd] + adst[(long)d * heads + hd]);
    atomicMaxF(&emax[(long)d * heads + hd], v);
}

// pass 2: segment sum of exp(e - emax)
__global__ void att_edge_sum(const int* __restrict__ src, const int* __restrict__ dst,
                             const float* __restrict__ asrc, const float* __restrict__ adst,
                             const float* __restrict__ emax, float* __restrict__ denom,
                             int E, int n, int heads) {
    long i = (long)blockIdx.x * blockDim.x + threadIdx.x;
    long total = (long)(E + n) * heads;
    if (i >= total) return;
    long e = i / heads; int hd = (int)(i % heads);
    int s, d;
    if (e < E) { s = src[e]; d = dst[e]; } else { s = d = (int)(e - E); }
    float v = leaky02(asrc[(long)s * heads + hd] + adst[(long)d * heads + hd]);
    atomicAdd(&denom[(long)d * heads + hd], expf(v - emax[(long)d * heads + hd]));
}

// pass 3: out[dst] += alpha * h[src]; one thread = (edge, 4 features)
__global__ void att_scatter(const int* __restrict__ src, const int* __restrict__ dst,
                            const float* __restrict__ asrc, const float* __restrict__ adst,
                            const float* __restrict__ emax, const float* __restrict__ denom,
                            const float* __restrict__ h, float* __restrict__ out,
                            int E, int n, int heads) {
    const int HD  = heads * HHID;
    const int tpe = HD / 4;
    long i = (long)blockIdx.x * blockDim.x + threadIdx.x;
    long total = (long)(E + n) * tpe;
    if (i >= total) return;
    long e = i / tpe;
    int  f4 = (int)(i % tpe);
    int  d0 = f4 * 4;
    int  hd = d0 / HHID;
    int s, d;
    if (e < E) { s = src[e]; d = dst[e]; } else { s = d = (int)(e - E); }
    float v = leaky02(asrc[(long)s * heads + hd] + adst[(long)d * heads + hd]);
    float alpha = expf(v - emax[(long)d * heads + hd]) / denom[(long)d * heads + hd];
    const float4 hv = *(const float4*)(h + (long)s * HD + d0);
    float* op = out + (long)d * HD + d0;
    atomicAdd(op + 0, hv.x * alpha);
    atomicAdd(op + 1, hv.y * alpha);
    atomicAdd(op + 2, hv.z * alpha);
    atomicAdd(op + 3, hv.w * alpha);
}

// in-place bias + ELU over [N, HD]
__global__ void bias_elu(float* __restrict__ v, const float* __restrict__ b, long n, int HD) {
    long i = (long)blockIdx.x * blockDim.x + threadIdx.x;
    if (i >= n * HD) return;
    float t = v[i] + b[i % HD];
    v[i] = t > 0.f ? t : (expf(t) - 1.f);
}

// ---------------- fused classifier: relu((agg2+b2)@Wc1+bc1)@Wc2+bc2 ----------------
__global__ void classifier_kernel(const float* __restrict__ agg2, const float* __restrict__ b2,
                                  const float* __restrict__ Wc1, const float* __restrict__ bc1,
                                  const float* __restrict__ Wc2, const float* __restrict__ bc2,
                                  float* __restrict__ out, int n) {
    __shared__ float w1s[64 * 16];
    __shared__ float w2s[32];
    __shared__ float b2s[64];
    __shared__ float bc1s[16];
    for (int i = threadIdx.x; i < 64 * 16; i += blockDim.x) w1s[i] = Wc1[i];
    if (threadIdx.x < 32) w2s[threadIdx.x]  = Wc2[threadIdx.x];
    if (threadIdx.x < 64) b2s[threadIdx.x]  = b2[threadIdx.x];
    if (threadIdx.x < 16) bc1s[threadIdx.x] = bc1[threadIdx.x];
    __syncthreads();

    int node = blockIdx.x * blockDim.x + threadIdx.x;
    if (node >= n) return;

    float xv[64];
    const float* ap = agg2 + (long)node * 64;
    #pragma unroll
    for (int k = 0; k < 64; ++k) xv[k] = ap[k] + b2s[k];

    float hb[16];
    #pragma unroll
    for (int j = 0; j < 16; ++j) hb[j] = bc1s[j];
    for (int k = 0; k < 64; ++k) {
        float v = xv[k];
        #pragma unroll
        for (int j = 0; j < 16; ++j) hb[j] += v * w1s[k * 16 + j];
    }
    float o0 = bc2[0], o1 = bc2[1];
    #pragma unroll
    for (int j = 0; j < 16; ++j) {
        float hv = hb[j] > 0.f ? hb[j] : 0.f;
        o0 += hv * w2s[j * 2];
        o1 += hv * w2s[j * 2 + 1];
    }
    out[(long)node * 2]     = o0;
    out[(long)node * 2 + 1] = o1;
}

// ---------------------------------------------------------------------------

static inline unsigned gridFor(long n, int b) { return (unsigned)((n + b - 1) / b); }

extern "C" void kernel_launch(void* const* d_in, const int* in_sizes, int n_in,
                              void* d_out, int out_size, void* d_ws, size_t ws_size,
                              hipStream_t stream) {
    const float* x      = (const float*)d_in[0];
    const int*   ei     = (const int*)  d_in[1];
    const float* gamma  = (const float*)d_in[6];
    const float* beta   = (const float*)d_in[7];
    const float* W1     = (const float*)d_in[8];
    const float* a_src1 = (const float*)d_in[9];
    const float* a_dst1 = (const float*)d_in[10];
    const float* b1     = (const float*)d_in[11];
    const float* W2     = (const float*)d_in[12];
    const float* a_src2 = (const float*)d_in[13];
    const float* a_dst2 = (const float*)d_in[14];
    const float* b2     = (const float*)d_in[15];
    const float* Wc1    = (const float*)d_in[16];
    const float* bc1    = (const float*)d_in[17];
    const float* Wc2    = (const float*)d_in[18];
    const float* bc2    = (const float*)d_in[19];
    float* out = (float*)d_out;

    const int N = in_sizes[0] / F_IN;   // 50000
    const int E = in_sizes[1] / 2;      // 800000
    const int* src = ei;
    const int* dst = ei + E;

    // workspace carve-up (floats)
    float* ws     = (float*)d_ws;
    float* sum    = ws;                      // 256
    float* sumsq  = ws + 256;                // 256
    float* sc     = ws + 512;                // 256
    float* sh     = ws + 768;                // 256
    float* h1     = ws + 1024;               // N*128
    float* agg1   = h1   + (long)N * 128;    // N*128 (becomes out1 in-place)
    float* h2     = agg1 + (long)N * 128;    // N*64
    float* agg2   = h2   + (long)N * 64;     // N*64
    float* asrc1  = agg2 + (long)N * 64;     // N*2
    float* adst1  = asrc1 + (long)N * 2;     // N*2
    float* emax1  = adst1 + (long)N * 2;     // N*2
    float* den1   = emax1 + (long)N * 2;     // N*2
    float* asrc2  = den1  + (long)N * 2;     // N
    float* adst2  = asrc2 + N;               // N
    float* emax2  = adst2 + N;               // N
    float* den2   = emax2 + N;               // N

    // ---- init (graph-safe, deterministic) ----
    fill_f32<<<gridFor(512, 256), 256, 0, stream>>>(sum, 512, 0.f);          // sum+sumsq
    fill_f32<<<gridFor((long)N * 128, 256), 256, 0, stream>>>(agg1, (long)N * 128, 0.f);
    fill_f32<<<gridFor((long)N * 64, 256), 256, 0, stream>>>(agg2, (long)N * 64, 0.f);
    fill_f32<<<gridFor((long)N * 2, 256), 256, 0, stream>>>(emax1, (long)N * 2, -1e30f);
    fill_f32<<<gridFor((long)N * 2, 256), 256, 0, stream>>>(den1, (long)N * 2, 0.f);
    fill_f32<<<gridFor(N, 256), 256, 0, stream>>>(emax2, N, -1e30f);
    fill_f32<<<gridFor(N, 256), 256, 0, stream>>>(den2, N, 0.f);

    // ---- BatchNorm stats -> per-feature scale/shift ----
    bn_stats<<<256, 256, 0, stream>>>(x, sum, sumsq, N);
    bn_finalize<<<1, 256, 0, stream>>>(sum, sumsq, gamma, beta, sc, sh, N);

    // ---- Layer 1: h1 = BN(x) @ W1  (WMMA, BN fused) ----
    gemm_wmma<256, 128><<<N / 16, 256, 0, stream>>>(x, W1, h1, sc, sh);
    att_dots<<<gridFor((long)N * 2, 256), 256, 0, stream>>>(h1, a_src1, a_dst1, asrc1, adst1, N, 2);

    long tot1 = (long)(E + N) * 2;
    att_edge_max<<<gridFor(tot1, 256), 256, 0, stream>>>(src, dst, asrc1, adst1, emax1, E, N, 2);
    att_edge_sum<<<gridFor(tot1, 256), 256, 0, stream>>>(src, dst, asrc1, adst1, emax1, den1, E, N, 2);
    long sc1 = (long)(E + N) * (128 / 4);
    att_scatter<<<gridFor(sc1, 256), 256, 0, stream>>>(src, dst, asrc1, adst1, emax1, den1,
                                                       h1, agg1, E, N, 2);
    bias_elu<<<gridFor((long)N * 128, 256), 256, 0, stream>>>(agg1, b1, N, 128);

    // ---- Layer 2: h2 = out1 @ W2 (WMMA), heads = 1 ----
    gemm_wmma<128, 64><<<N / 16, 128, 0, stream>>>(agg1, W2, h2, nullptr, nullptr);
    att_dots<<<gridFor(N, 256), 256, 0, stream>>>(h2, a_src2, a_dst2, asrc2, adst2, N, 1);

    long tot2 = (long)(E + N);
    att_edge_max<<<gridFor(tot2, 256), 256, 0, stream>>>(src, dst, asrc2, adst2, emax2, E, N, 1);
    att_edge_sum<<<gridFor(tot2, 256), 256, 0, stream>>>(src, dst, asrc2, adst2, emax2, den2, E, N, 1);
    long sc2 = (long)(E + N) * (64 / 4);
    att_scatter<<<gridFor(sc2, 256), 256, 0, stream>>>(src, dst, asrc2, adst2, emax2, den2,
                                                       h2, agg2, E, N, 1);

    // ---- classifier -> d_out [N,2] ----
    classifier_kernel<<<gridFor(N, 256), 256, 0, stream>>>(agg2, b2, Wc1, bc1, Wc2, bc2, out, N);
}